// GraphLearning_29652454211798
// MI455X (gfx1250) — compile-verified
//
#include <hip/hip_runtime.h>
#include <hip/hip_bf16.h>
#include <math.h>

// ---------------- problem constants (match reference) ----------------
#define NNODE   100000
#define NEDGE   1600000
#define TOTE    (NEDGE + NNODE)      // edges + self loops
#define FIN     128
#define CDIM    16
#define HEADS   2
#define HC      32                   // HEADS*CDIM
#define EDIM    4
#define NLAYER  2
#define EPSV    1e-5f
#define SLOPE   0.2f
#define NTILES  (NNODE / 16)         // 6250, exact

typedef __attribute__((ext_vector_type(2))) float v2f;
typedef __attribute__((ext_vector_type(8))) float v8f;

// D = A(16x4 f32) * B(4x16 f32) + C  — full f32 WMMA on CDNA5
__device__ __forceinline__ v8f wmma4(v2f a, v2f b, v8f c) {
  return __builtin_amdgcn_wmma_f32_16x16x4_f32(false, a, false, b, (short)0, c,
                                               false, false);
}

// order-preserving float<->uint keys for atomic max
__device__ __forceinline__ unsigned fkey(float f) {
  unsigned u = __float_as_uint(f);
  return (u & 0x80000000u) ? ~u : (u | 0x80000000u);
}
__device__ __forceinline__ float fkey_dec(unsigned k) {
  return (k & 0x80000000u) ? __uint_as_float(k & 0x7FFFFFFFu)
                           : __uint_as_float(~k);
}
#define KEY_NEGINF 0x007FFFFFu   // fkey(-inf)

// ================= embed GEMM: x[N,128] @ W[128,16] ==================
__global__ void k_embed_gemm(const float* __restrict__ x,
                             const float* __restrict__ W,
                             const float* __restrict__ bias,
                             float* __restrict__ hraw) {
  int wave = (int)((blockIdx.x * blockDim.x + threadIdx.x) >> 5);
  if (wave >= NTILES) return;
  int lane = threadIdx.x & 31, half = lane >> 4, idx = lane & 15;
  int row0 = wave * 16, r = row0 + idx;
  const float2* x2 = (const float2*)x;
  v8f acc = {};
  for (int k0 = 0; k0 < FIN; k0 += 4) {
    int k = k0 + half * 2;
    float2 av = x2[(size_t)r * (FIN / 2) + (k >> 1)];
    v2f a; a.x = av.x; a.y = av.y;
    v2f b; b.x = W[k * CDIM + idx]; b.y = W[(k + 1) * CDIM + idx];
    acc = wmma4(a, b, acc);
  }
  float bv = bias[idx];
  for (int v = 0; v < 8; v++)
    hraw[(size_t)(row0 + v + half * 8) * CDIM + idx] = acc[v] + bv;
}

// ============ per-node LayerNorm(16) + ReLU ; seed h and JK-max out ==
__global__ void k_embed_ln(const float* __restrict__ raw,
                           const float* __restrict__ w,
                           const float* __restrict__ b,
                           float* __restrict__ h, float* __restrict__ out) {
  int n = blockIdx.x * blockDim.x + threadIdx.x;
  if (n >= NNODE) return;
  float v[CDIM], mu = 0.f;
  for (int c = 0; c < CDIM; c++) { v[c] = raw[(size_t)n * CDIM + c]; mu += v[c]; }
  mu *= (1.0f / CDIM);
  float var = 0.f;
  for (int c = 0; c < CDIM; c++) { float d = v[c] - mu; var += d * d; }
  var *= (1.0f / CDIM);
  float rstd = rsqrtf(var + EPSV);
  for (int c = 0; c < CDIM; c++) {
    float o = fmaxf((v[c] - mu) * rstd * w[c] + b[c], 0.f);
    h[(size_t)n * CDIM + c] = o;
    out[(size_t)n * CDIM + c] = o;
  }
}

// ======== xl = h@Wl+bl, xr = h@Wr+br (both [N,32]) via WMMA ==========
__global__ void k_gemm_xlxr(const float* __restrict__ h,
                            const float* __restrict__ Wl, const float* __restrict__ bl,
                            const float* __restrict__ Wr, const float* __restrict__ br,
                            float* __restrict__ xl, float* __restrict__ xr) {
  int wave = (int)((blockIdx.x * blockDim.x + threadIdx.x) >> 5);
  if (wave >= NTILES) return;
  int lane = threadIdx.x & 31, half = lane >> 4, idx = lane & 15;
  int row0 = wave * 16, r = row0 + idx;
  const float2* h2 = (const float2*)h;
  v8f aL0 = {}, aL1 = {}, aR0 = {}, aR1 = {};
  for (int k0 = 0; k0 < CDIM; k0 += 4) {
    int k = k0 + half * 2;
    float2 av = h2[(size_t)r * (CDIM / 2) + (k >> 1)];
    v2f a; a.x = av.x; a.y = av.y;
    v2f b;
    b.x = Wl[k * HC + idx];      b.y = Wl[(k + 1) * HC + idx];      aL0 = wmma4(a, b, aL0);
    b.x = Wl[k * HC + 16 + idx]; b.y = Wl[(k + 1) * HC + 16 + idx]; aL1 = wmma4(a, b, aL1);
    b.x = Wr[k * HC + idx];      b.y = Wr[(k + 1) * HC + idx];      aR0 = wmma4(a, b, aR0);
    b.x = Wr[k * HC + 16 + idx]; b.y = Wr[(k + 1) * HC + 16 + idx]; aR1 = wmma4(a, b, aR1);
  }
  float bL0 = bl[idx], bL1 = bl[16 + idx], bR0 = br[idx], bR1 = br[16 + idx];
  for (int v = 0; v < 8; v++) {
    size_t m = (size_t)(row0 + v + half * 8) * HC;
    xl[m + idx]      = aL0[v] + bL0;
    xl[m + 16 + idx] = aL1[v] + bL1;
    xr[m + idx]      = aR0[v] + bR0;
    xr[m + 16 + idx] = aR1[v] + bR1;
  }
}

// ============ per-layer scratch init ================================
__global__ void k_init_layer(float* __restrict__ g, unsigned* __restrict__ mkey,
                             float* __restrict__ den, double* __restrict__ acc) {
  size_t t = (size_t)blockIdx.x * blockDim.x + threadIdx.x;
  if (t < (size_t)NNODE * HC) g[t] = 0.f;
  if (t < (size_t)NNODE * HEADS) { mkey[t] = KEY_NEGINF; den[t] = 0.f; }
  if (t < 4) acc[t] = 0.0;
}

// ============ edge pass 1: logits + segment max over dst ============
__global__ void k_edge_logits(const int* __restrict__ src, const int* __restrict__ dst,
                              const float* __restrict__ ea,
                              const float* __restrict__ xl, const float* __restrict__ xr,
                              const float* __restrict__ We, const float* __restrict__ att,
                              float* __restrict__ logits, unsigned* __restrict__ mkey) {
  __shared__ float sWe[EDIM * HC];
  __shared__ float sAtt[HC];
  int t = threadIdx.x;
  if (t < EDIM * HC) sWe[t] = We[t];
  if (t >= 128 && t < 128 + HC) sAtt[t - 128] = att[t - 128];
  __syncthreads();
  size_t e = (size_t)blockIdx.x * blockDim.x + t;
  if (e >= (size_t)TOTE) return;

  int s, d; float w0 = 0.f, w1 = 0.f, w2 = 0.f, w3 = 0.f;
  bool real = e < (size_t)NEDGE;
  if (real) {
    s = src[e]; d = dst[e];
    float4 wa = ((const float4*)ea)[e];
    w0 = wa.x; w1 = wa.y; w2 = wa.z; w3 = wa.w;
  } else {
    s = d = (int)(e - NEDGE);
  }
  const float4* a4 = (const float4*)(xl + (size_t)s * HC);
  const float4* b4 = (const float4*)(xr + (size_t)d * HC);
  float msg[HC];
  for (int j = 0; j < 8; j++) {
    float4 va = a4[j], vb = b4[j];
    int c = 4 * j;
    msg[c + 0] = va.x + vb.x; msg[c + 1] = va.y + vb.y;
    msg[c + 2] = va.z + vb.z; msg[c + 3] = va.w + vb.w;
  }
  if (real)
    for (int c = 0; c < HC; c++)
      msg[c] += w0 * sWe[0 * HC + c] + w1 * sWe[1 * HC + c] +
                w2 * sWe[2 * HC + c] + w3 * sWe[3 * HC + c];
  for (int hh = 0; hh < HEADS; hh++) {
    float lg = 0.f;
    for (int c = 0; c < CDIM; c++) {
      float mv = msg[hh * CDIM + c];
      float lr = mv > 0.f ? mv : SLOPE * mv;
      lg += lr * sAtt[hh * CDIM + c];
    }
    logits[e * HEADS + hh] = lg;
    atomicMax(&mkey[(size_t)d * HEADS + hh], fkey(lg));
  }
}

// ==== edge pass 2: ex=exp(l-m); den += ex; g[dst] += ex * xl[src] ====
__global__ void k_edge_accum(const int* __restrict__ src, const int* __restrict__ dst,
                             const float* __restrict__ xl,
                             const float* __restrict__ logits,
                             const unsigned* __restrict__ mkey,
                             float* __restrict__ den, float* __restrict__ g) {
  size_t e = (size_t)blockIdx.x * blockDim.x + threadIdx.x;
  if (e >= (size_t)TOTE) return;
  int s, d;
  if (e < (size_t)NEDGE) { s = src[e]; d = dst[e]; }
  else                   { s = d = (int)(e - NEDGE); }
  float ex[HEADS];
  for (int hh = 0; hh < HEADS; hh++) {
    float m = fkey_dec(mkey[(size_t)d * HEADS + hh]);
    ex[hh] = __expf(logits[e * HEADS + hh] - m);
    atomicAdd(&den[(size_t)d * HEADS + hh], ex[hh]);
  }
  const float4* a4 = (const float4*)(xl + (size_t)s * HC);
  float* gd = g + (size_t)d * HC;
  for (int j = 0; j < 8; j++) {
    float4 va = a4[j];
    float w = ex[j >> 2];                // head = (4j)/16
    int c = 4 * j;
    atomicAdd(&gd[c + 0], w * va.x);
    atomicAdd(&gd[c + 1], w * va.y);
    atomicAdd(&gd[c + 2], w * va.z);
    atomicAdd(&gd[c + 3], w * va.w);
  }
}

// ==== node finish: g = g/den + gat_b ; accumulate graph-LN1 stats ====
__global__ void k_node_finish(float* __restrict__ g, const float* __restrict__ den,
                              const float* __restrict__ gat_b,
                              double* __restrict__ acc) {
  __shared__ float s1[256], s2[256];
  size_t t = (size_t)blockIdx.x * blockDim.x + threadIdx.x;
  float v = 0.f;
  if (t < (size_t)NNODE * HC) {
    size_t node = t >> 5; int c = (int)(t & 31); int hh = c >> 4;
    v = g[t] / den[node * HEADS + hh] + gat_b[c];
    g[t] = v;
  }
  int tid = threadIdx.x;
  s1[tid] = v; s2[tid] = v * v;
  __syncthreads();
  for (int o = 128; o > 0; o >>= 1) {
    if (tid < o) { s1[tid] += s1[tid + o]; s2[tid] += s2[tid + o]; }
    __syncthreads();
  }
  if (tid == 0) { atomicAdd(&acc[0], (double)s1[0]); atomicAdd(&acc[1], (double)s2[0]); }
}

__global__ void k_finalize_stats(const double* __restrict__ acc,
                                 float* __restrict__ st, double count) {
  if (threadIdx.x == 0 && blockIdx.x == 0) {
    double mu = acc[0] / count;
    double var = acc[1] / count - mu * mu;
    st[0] = (float)mu;
    st[1] = (float)(1.0 / sqrt(var + (double)EPSV));
  }
}

// ==== lin GEMM: y = relu(graphLN1(g)) @ lin_W + lin_b ; LN2 stats ====
__global__ void k_gemm_lin(const float* __restrict__ g, const float* __restrict__ st,
                           const float* __restrict__ n1w, const float* __restrict__ n1b,
                           const float* __restrict__ linW, const float* __restrict__ linb,
                           float* __restrict__ y, double* __restrict__ acc) {
  __shared__ float s1[128], s2[128];
  int wave = (int)((blockIdx.x * blockDim.x + threadIdx.x) >> 5);
  int lane = threadIdx.x & 31, half = lane >> 4, idx = lane & 15;
  bool active = wave < NTILES;
  float mu = st[0], rstd = st[1];
  float ls = 0.f, ls2 = 0.f;
  if (active) {
    int row0 = wave * 16, r = row0 + idx;
    v8f acc8 = {};
    for (int k0 = 0; k0 < HC; k0 += 4) {
      int k = k0 + half * 2;
      float g0 = g[(size_t)r * HC + k], g1 = g[(size_t)r * HC + k + 1];
      v2f a;
      a.x = fmaxf((g0 - mu) * rstd * n1w[k] + n1b[k], 0.f);
      a.y = fmaxf((g1 - mu) * rstd * n1w[k + 1] + n1b[k + 1], 0.f);
      v2f b; b.x = linW[k * CDIM + idx]; b.y = linW[(k + 1) * CDIM + idx];
      acc8 = wmma4(a, b, acc8);
    }
    float bias = linb[idx];
    for (int v = 0; v < 8; v++) {
      float yv = acc8[v] + bias;
      y[(size_t)(row0 + v + half * 8) * CDIM + idx] = yv;
      ls += yv; ls2 += yv * yv;
    }
  }
  int tid = threadIdx.x;
  s1[tid] = ls; s2[tid] = ls2;
  __syncthreads();
  for (int o = 64; o > 0; o >>= 1) {
    if (tid < o) { s1[tid] += s1[tid + o]; s2[tid] += s2[tid + o]; }
    __syncthreads();
  }
  if (tid == 0) { atomicAdd(&acc[2], (double)s1[0]); atomicAdd(&acc[3], (double)s2[0]); }
}

// ==== graphLN2 + residual + ReLU ; h update ; JK elementwise max =====
__global__ void k_ln2_res(const float* __restrict__ y, const float* __restrict__ st2,
                          const float* __restrict__ n2w, const float* __restrict__ n2b,
                          float* __restrict__ h, float* __restrict__ out) {
  size_t t = (size_t)blockIdx.x * blockDim.x + threadIdx.x;
  if (t >= (size_t)NNODE * CDIM) return;
  int c = (int)(t & 15);
  float mu = st2[0], rstd = st2[1];
  float yv = (y[t] - mu) * rstd * n2w[c] + n2b[c];
  float hn = fmaxf(yv + h[t], 0.f);
  h[t] = hn;
  out[t] = fmaxf(out[t], hn);
}

// ====================================================================
extern "C" void kernel_launch(void* const* d_in, const int* in_sizes, int n_in,
                              void* d_out, int out_size, void* d_ws, size_t ws_size,
                              hipStream_t stream) {
  const float* x       = (const float*)d_in[0];
  const int*   ei      = (const int*)d_in[1];   // [2,E] int (per harness dtype map)
  const float* eattr   = (const float*)d_in[2];
  const float* embed_W = (const float*)d_in[3];
  const float* embed_b = (const float*)d_in[4];
  const float* ln0_w   = (const float*)d_in[5];
  const float* ln0_b   = (const float*)d_in[6];
  const float* Wl      = (const float*)d_in[7];
  const float* bl      = (const float*)d_in[8];
  const float* Wr      = (const float*)d_in[9];
  const float* br      = (const float*)d_in[10];
  const float* We      = (const float*)d_in[11];
  const float* att     = (const float*)d_in[12];
  const float* gat_b   = (const float*)d_in[13];
  const float* n1_w    = (const float*)d_in[14];
  const float* n1_b    = (const float*)d_in[15];
  const float* lin_W   = (const float*)d_in[16];
  const float* lin_b   = (const float*)d_in[17];
  const float* n2_w    = (const float*)d_in[18];
  const float* n2_b    = (const float*)d_in[19];
  float* out = (float*)d_out;

  const int* srcp = ei;
  const int* dstp = ei + NEDGE;

  // -------- workspace carve-up (all chunks 256B aligned) --------
  char* w = (char*)d_ws;
  size_t off = 0;
  float*    h      = (float*)(w + off); off += (size_t)NNODE * CDIM  * 4;  // 6.4MB
  float*    xl     = (float*)(w + off); off += (size_t)NNODE * HC    * 4;  // 12.8MB
  float*    xr     = (float*)(w + off); off += (size_t)NNODE * HC    * 4;  // 12.8MB
  float*    g      = (float*)(w + off); off += (size_t)NNODE * HC    * 4;  // 12.8MB
  unsigned* mkey   = (unsigned*)(w + off); off += (size_t)NNODE * HEADS * 4;
  float*    den    = (float*)(w + off); off += (size_t)NNODE * HEADS * 4;
  float*    logits = (float*)(w + off); off += (size_t)TOTE  * HEADS * 4;  // 13.6MB
  float*    ybuf   = (float*)(w + off); off += (size_t)NNODE * CDIM  * 4;  // 6.4MB
  double*   acc    = (double*)(w + off); off += 256;                        // 4 doubles
  float*    stats  = (float*)(w + off); off += 256;                         // 4 floats
  (void)ws_size; (void)n_in; (void)in_sizes; (void)out_size;

  const int TB = 256;
  dim3 gemmGrid((NTILES + 3) / 4), gemmBlk(128);          // 4 waves / block
  dim3 edgeGrid((TOTE + TB - 1) / TB), blk(TB);
  dim3 nodeGrid((NNODE + TB - 1) / TB);
  dim3 ncGrid(((size_t)NNODE * HC + TB - 1) / TB);        // 12500 exact
  dim3 ocGrid(((size_t)NNODE * CDIM + TB - 1) / TB);

  // ---- embedding: GEMM (WMMA) -> per-node LN -> ReLU ; seed out ----
  k_embed_gemm<<<gemmGrid, gemmBlk, 0, stream>>>(x, embed_W, embed_b, ybuf);
  k_embed_ln<<<nodeGrid, blk, 0, stream>>>(ybuf, ln0_w, ln0_b, h, out);

  for (int l = 0; l < NLAYER; l++) {
    const float* Wl_l  = Wl    + (size_t)l * CDIM * HC;
    const float* bl_l  = bl    + (size_t)l * HC;
    const float* Wr_l  = Wr    + (size_t)l * CDIM * HC;
    const float* br_l  = br    + (size_t)l * HC;
    const float* We_l  = We    + (size_t)l * EDIM * HC;
    const float* att_l = att   + (size_t)l * HEADS * CDIM;
    const float* gb_l  = gat_b + (size_t)l * HC;
    const float* n1w_l = n1_w  + (size_t)l * HC;
    const float* n1b_l = n1_b  + (size_t)l * HC;
    const float* lW_l  = lin_W + (size_t)l * HC * CDIM;
    const float* lb_l  = lin_b + (size_t)l * CDIM;
    const float* n2w_l = n2_w  + (size_t)l * CDIM;
    const float* n2b_l = n2_b  + (size_t)l * CDIM;

    k_gemm_xlxr<<<gemmGrid, gemmBlk, 0, stream>>>(h, Wl_l, bl_l, Wr_l, br_l, xl, xr);
    k_init_layer<<<ncGrid, blk, 0, stream>>>(g, mkey, den, acc);
    k_edge_logits<<<edgeGrid, blk, 0, stream>>>(srcp, dstp, eattr, xl, xr,
                                                We_l, att_l, logits, mkey);
    k_edge_accum<<<edgeGrid, blk, 0, stream>>>(srcp, dstp, xl, logits, mkey, den, g);
    k_node_finish<<<ncGrid, blk, 0, stream>>>(g, den, gb_l, acc);
    k_finalize_stats<<<1, 32, 0, stream>>>(acc, stats, (double)NNODE * HC);
    k_gemm_lin<<<gemmGrid, gemmBlk, 0, stream>>>(g, stats, n1w_l, n1b_l,
                                                 lW_l, lb_l, ybuf, acc);
    k_finalize_stats<<<1, 32, 0, stream>>>(acc + 2, stats + 2, (double)NNODE * CDIM);
    k_ln2_res<<<ocGrid, blk, 0, stream>>>(ybuf, stats + 2, n2w_l, n2b_l, h, out);
  }
}